// Model_17188459119206
// MI455X (gfx1250) — compile-verified
//
#include <hip/hip_runtime.h>
#include <hip/hip_bf16.h>

// CDNA5 (gfx1250) MLP 2->64->64->64->1 with chaotic activation.
// GEMMs: exact-fp32 WMMA (v_wmma_f32_16x16x4_f32), wave-private 16-row tiles,
// activations through bank-conflict-free padded LDS.
// Activation cos uses the HW transcendental (v_cos_f32): pre-activation values
// are bounded (|z| <~ 10), so no Payne-Hanek reduction is ever needed and the
// ~1ulp native accuracy matches fp32 matmul rounding noise.

typedef __attribute__((ext_vector_type(2))) float v2f;
typedef __attribute__((ext_vector_type(8))) float v8f;

#define R_CONST 3.82843f
#define LDSH 68   // padded row stride (floats): banks = 4*row + k (conflict-free b64 loads)

__device__ __forceinline__ float act(float x) {
    // cos squash into [0,1], then two logistic-map iterations (match ref order)
    x = __cosf(x) * 0.5f + 0.5f;              // v_mul (1/2pi) + v_cos_f32
    x = R_CONST * x * (1.0f - x);
    x = R_CONST * x * (1.0f - x);
    return x;
}

__device__ __forceinline__ v8f wmma_f32(v2f a, v2f b, v8f c) {
    // 8 args: (neg_a, A, neg_b, B, c_mod, C, reuse_a, reuse_b)
    return __builtin_amdgcn_wmma_f32_16x16x4_f32(false, a, false, b, (short)0, c,
                                                 false, false);
}

// One 64->64 dense layer + activation for this wave's 16-row tile.
__device__ __forceinline__ void dense64(float* __restrict__ h,
                                        const float* __restrict__ sW,
                                        const float* __restrict__ biasS,
                                        int row, int hi) {
    // Preload all 16 A-fragments (each reused across 4 N-tiles).
    v2f afr[16];
#pragma unroll
    for (int k = 0; k < 16; ++k) {
        afr[k] = *(const v2f*)&h[row * LDSH + 4 * k + 2 * hi];
    }
#pragma unroll
    for (int n = 0; n < 4; ++n) {
        float bv = biasS[16 * n + row];           // bias depends only on N = lane&15
        v8f acc = {bv, bv, bv, bv, bv, bv, bv, bv};
#pragma unroll
        for (int k = 0; k < 16; ++k) {
            // B(K=4k+2*hi, N=16n+row) = W[16n+row][4k+2*hi]  (since h@W^T)
            v2f bfr = *(const v2f*)&sW[(16 * n + row) * LDSH + 4 * k + 2 * hi];
            acc = wmma_f32(afr[k], bfr, acc);
        }
#pragma unroll
        for (int r = 0; r < 8; ++r) {
            int M = r + 8 * hi;                   // C/D layout: lanes 16-31 hold M=r+8
            h[M * LDSH + 16 * n + row] = act(acc[r]);
        }
    }
}

__global__ __launch_bounds__(256) void mlp_mlp4_kernel(
    const float* __restrict__ x,
    const float* __restrict__ W1, const float* __restrict__ b1,
    const float* __restrict__ W2, const float* __restrict__ b2,
    const float* __restrict__ W3, const float* __restrict__ b3,
    const float* __restrict__ W4, const float* __restrict__ b4,
    float* __restrict__ out, int N) {

    __shared__ float sW[64 * LDSH];          // W2 then W3 (reloaded, barriered)
    __shared__ float sH[8 * 16 * LDSH];      // per-wave activation tiles
    __shared__ float sW1[64 * 2];
    __shared__ float sB1[64], sB2[64], sB3[64], sW4[64];

    const int tid  = threadIdx.x;
    const int wave = tid >> 5;
    const int lane = tid & 31;
    const int row  = lane & 15;
    const int hi   = lane >> 4;
    const float loMask = (hi == 0) ? 1.0f : 0.0f;   // zero K=2,3 contributions

    // ---- cooperative loads: W2 (padded) + small tensors ----
    for (int i = tid; i < 64 * 64; i += 256) {
        int n = i >> 6, k = i & 63;
        sW[n * LDSH + k] = W2[i];
    }
    if (tid < 128) sW1[tid] = W1[tid];
    if (tid < 64) {
        sB1[tid] = b1[tid];
        sB2[tid] = b2[tid];
        sB3[tid] = b3[tid];
        sW4[tid] = W4[tid];
    }
    __syncthreads();

    const long base = (long)blockIdx.x * 128 + (long)wave * 16;
    float* h = &sH[wave * 16 * LDSH];

    // ---- Layer 1: h = act(x @ W1^T + b1), K=2 zero-padded to 4 (exact) ----
    long rclamp = base + row;
    if (rclamp >= N) rclamp = N - 1;             // safe tail (N%128==0 in practice)
    {
        v2f a1;
        float x0 = x[rclamp * 2 + 0];
        float x1 = x[rclamp * 2 + 1];
        a1.x = x0 * loMask;                      // force unconditional load + mask
        a1.y = x1 * loMask;
#pragma unroll
        for (int n = 0; n < 4; ++n) {
            int wn = 16 * n + row;
            float w0 = sW1[wn * 2 + 0];          // uniform-EXEC ds_load
            float w1 = sW1[wn * 2 + 1];
            v2f bfr;
            bfr.x = w0 * loMask;
            bfr.y = w1 * loMask;
            float bv = sB1[16 * n + row];
            v8f acc = {bv, bv, bv, bv, bv, bv, bv, bv};
            acc = wmma_f32(a1, bfr, acc);
#pragma unroll
            for (int r = 0; r < 8; ++r) {
                int M = r + 8 * hi;
                h[M * LDSH + 16 * n + row] = act(acc[r]);
            }
        }
    }

    // ---- Layer 2 ----
    dense64(h, sW, sB2, row, hi);

    // ---- swap W3 into the shared weight buffer ----
    __syncthreads();
    for (int i = tid; i < 64 * 64; i += 256) {
        int n = i >> 6, k = i & 63;
        sW[n * LDSH + k] = W3[i];
    }
    __syncthreads();

    // ---- Layer 3 ----
    dense64(h, sW, sB3, row, hi);

    // ---- Layer 4: out = act(h . W4 + b4), split dot across lane halves ----
    float partial = 0.0f;
    const float* hr = &h[row * LDSH + hi * 32];   // 8B-aligned (68*row + 32*hi even)
    const float* wr = &sW4[hi * 32];
#pragma unroll
    for (int k = 0; k < 16; ++k) {
        v2f hv = *(const v2f*)&hr[2 * k];
        v2f wv = *(const v2f*)&wr[2 * k];
        partial += hv.x * wv.x;
        partial += hv.y * wv.y;
    }
    partial += __shfl_xor(partial, 16, 32);
    if (lane < 16 && (base + row) < N) {
        out[base + row] = act(partial + b4[0]);
    }
}

extern "C" void kernel_launch(void* const* d_in, const int* in_sizes, int n_in,
                              void* d_out, int out_size, void* d_ws, size_t ws_size,
                              hipStream_t stream) {
    (void)n_in; (void)d_ws; (void)ws_size; (void)out_size;
    const float* x  = (const float*)d_in[0];
    const float* W1 = (const float*)d_in[1];
    const float* b1 = (const float*)d_in[2];
    const float* W2 = (const float*)d_in[3];
    const float* b2 = (const float*)d_in[4];
    const float* W3 = (const float*)d_in[5];
    const float* b3 = (const float*)d_in[6];
    const float* W4 = (const float*)d_in[7];
    const float* b4 = (const float*)d_in[8];
    float* out = (float*)d_out;

    int N = in_sizes[0] / 2;                 // x is [N,2]
    int blocks = (N + 127) / 128;            // 128 rows per block (8 waves x 16 rows)
    mlp_mlp4_kernel<<<blocks, 256, 0, stream>>>(x, W1, b1, W2, b2, W3, b3, W4, b4,
                                                out, N);
}